// AttentionBasedSummarizer_40802189312828
// MI455X (gfx1250) — compile-verified
//
#include <hip/hip_runtime.h>
#include <hip/hip_bf16.h>
#include <cstdint>

#define B_  8
#define T_  2048
#define D_  128
#define NC_ 32   // T chunks for the weighted-sum partial reduction

typedef __attribute__((ext_vector_type(2))) float v2f;
typedef __attribute__((ext_vector_type(8))) float v8f;

// ---------------------------------------------------------------------------
// Kernel 1: s[b,t] = dot(H[b,t,0:128], w[0:128]) + bias
// One wave32 per 16 rows. Uses V_WMMA_F32_16X16X4_F32:
//   A (16x4): row 0 = w chunk  (lane0 holds K=0,1; lane16 holds K=2,3)
//   B (4x16): H tile, B[k][n] = H[b, t0+n, dk+k]
//   D row 0, col n  ==  partial dot for row t0+n   (lanes 0..15, element 0)
// ---------------------------------------------------------------------------
__global__ __launch_bounds__(32)
void k_scores(const float* __restrict__ H, const float* __restrict__ w,
              const float* __restrict__ wb, float* __restrict__ s) {
  const int lane = threadIdx.x;      // 0..31
  const int col  = lane & 15;        // N index (t within tile)
  const int hi   = lane >> 4;        // 0 -> K=0,1 ; 1 -> K=2,3
  const int blk  = blockIdx.x;       // B*(T/16) = 1024
  const int b    = blk >> 7;         // / 128
  const int t0   = (blk & 127) << 4;

  const float* rowp = H + ((size_t)b * T_ + (size_t)(t0 + col)) * D_ + hi * 2;

  v8f acc = {};
  for (int dk = 0; dk < D_; dk += 4) {
    // A: only matrix row 0 is nonzero (w values); other rows zero.
    const float2 wv = *(const float2*)(w + dk + hi * 2);
    v2f a;
    a.x = (col == 0) ? wv.x : 0.0f;
    a.y = (col == 0) ? wv.y : 0.0f;
    // B: lane<16 holds K=dk+0,dk+1 of column `col`; lane>=16 holds K=dk+2,dk+3.
    const float2 hv = *(const float2*)(rowp + dk);
    v2f bm;
    bm.x = hv.x;
    bm.y = hv.y;
    acc = __builtin_amdgcn_wmma_f32_16x16x4_f32(false, a, false, bm,
                                                (short)0, acc, false, false);
  }
  if (lane < 16) {
    s[(size_t)b * T_ + t0 + lane] = acc[0] + wb[0];
  }
}

// ---------------------------------------------------------------------------
// Kernel 2: in-place softmax over T per batch: s[b,:] -> p[b,:]
// ---------------------------------------------------------------------------
__global__ __launch_bounds__(256)
void k_softmax(float* __restrict__ s) {
  const int b   = blockIdx.x;
  const int tid = threadIdx.x;
  float* sb = s + (size_t)b * T_;
  __shared__ float red[256];

  float m = -3.4e38f;
  for (int j = tid; j < T_; j += 256) m = fmaxf(m, sb[j]);
  red[tid] = m;
  __syncthreads();
  for (int off = 128; off > 0; off >>= 1) {
    if (tid < off) red[tid] = fmaxf(red[tid], red[tid + off]);
    __syncthreads();
  }
  m = red[0];
  __syncthreads();

  float l = 0.0f;
  for (int j = tid; j < T_; j += 256) {
    const float e = __expf(sb[j] - m);
    sb[j] = e;
    l += e;
  }
  red[tid] = l;
  __syncthreads();
  for (int off = 128; off > 0; off >>= 1) {
    if (tid < off) red[tid] += red[tid + off];
    __syncthreads();
  }
  const float inv = 1.0f / red[0];
  for (int j = tid; j < T_; j += 256) sb[j] *= inv;
}

// ---------------------------------------------------------------------------
// Kernel 3: partial weighted sums  partial[b,c,d] = sum_{j in chunk c} p[b,j]*H[b,j,d]
// grid (B, NC), 128 threads (one per d). Coalesced 512B row reads.
// ---------------------------------------------------------------------------
__global__ __launch_bounds__(D_)
void k_wsum_partial(const float* __restrict__ H, const float* __restrict__ p,
                    float* __restrict__ partial) {
  const int b = blockIdx.x;
  const int c = blockIdx.y;
  const int d = threadIdx.x;
  const int RC = T_ / NC_;           // 64 rows per chunk
  const int j0 = c * RC;

  const float* Hb = H + ((size_t)b * T_ + j0) * D_ + d;
  const float* pb = p + (size_t)b * T_ + j0;

  float acc = 0.0f;
  for (int j = 0; j < RC; ++j) {
    __builtin_prefetch(Hb + (size_t)(j + 8) * D_, 0, 0);   // global_prefetch_b8
    acc += pb[j] * Hb[(size_t)j * D_];
  }
  partial[((size_t)b * NC_ + c) * D_ + d] = acc;
}

// ---------------------------------------------------------------------------
// Kernel 4: r[b,d] = sum_c partial[b,c,d]
// ---------------------------------------------------------------------------
__global__ __launch_bounds__(D_)
void k_reduce(const float* __restrict__ partial, float* __restrict__ r) {
  const int b = blockIdx.x;
  const int d = threadIdx.x;
  float acc = 0.0f;
  for (int c = 0; c < NC_; ++c) acc += partial[((size_t)b * NC_ + c) * D_ + d];
  r[b * D_ + d] = acc;
}

// ---------------------------------------------------------------------------
// Kernel 5: broadcast r[b,:] to out[b, 0..T, :]  (float4 stores, 8 MB)
// ---------------------------------------------------------------------------
__global__ __launch_bounds__(256)
void k_broadcast(const float* __restrict__ r, float4* __restrict__ out) {
  const size_t i   = (size_t)blockIdx.x * 256 + threadIdx.x; // < B*T*D/4
  const int perb   = T_ * D_ / 4;                            // 65536
  const int b      = (int)(i / perb);
  const int d4     = (int)(i & (D_ / 4 - 1));
  out[i] = ((const float4*)(r + b * D_))[d4];
}

// ---------------------------------------------------------------------------
extern "C" void kernel_launch(void* const* d_in, const int* in_sizes, int n_in,
                              void* d_out, int out_size, void* d_ws, size_t ws_size,
                              hipStream_t stream) {
  (void)in_sizes; (void)n_in; (void)out_size; (void)ws_size;

  const float* H  = (const float*)d_in[0];   // (B,T,D) f32
  const float* w  = (const float*)d_in[1];   // (1,D) f32
  const float* wb = (const float*)d_in[2];   // (1,)    f32
  float* out = (float*)d_out;

  char* ws = (char*)d_ws;
  float* s       = (float*)ws;                                   // B*T      (s -> p in place)
  float* partial = (float*)(ws + (size_t)B_ * T_ * 4);           // B*NC*D
  float* r       = (float*)(ws + (size_t)B_ * T_ * 4
                               + (size_t)B_ * NC_ * D_ * 4);     // B*D

  k_scores      <<<B_ * (T_ / 16), 32,  0, stream>>>(H, w, wb, s);
  k_softmax     <<<B_,             256, 0, stream>>>(s);
  k_wsum_partial<<<dim3(B_, NC_),  D_,  0, stream>>>(H, s, partial);
  k_reduce      <<<B_,             D_,  0, stream>>>(partial, r);
  k_broadcast   <<<(B_ * T_ * D_ / 4) / 256, 256, 0, stream>>>(r, (float4*)out);
}